// DistOA_75015898792311
// MI455X (gfx1250) — compile-verified
//
#include <hip/hip_runtime.h>
#include <hip/hip_bf16.h>

typedef __attribute__((ext_vector_type(2))) float v2f;
typedef __attribute__((ext_vector_type(8))) float v8f;

#define NB   256
#define PB   256
#define NPP  512
#define SS   260   // LDS row stride (floats), padded to dodge bank conflicts

__device__ __forceinline__ float sigmoidf_(float x) {
    return 1.0f / (1.0f + __expf(-x));
}

// One workgroup per (k,b): build Cm in LDS + b/rowsum in ws, LU-factorize with
// WMMA trailing updates, then per-wave triangular solves with WMMA block updates.
// Convention: strictly-below-diagonal entries of the factored matrix hold -L
// (negated Doolittle multipliers) so WMMA "A*B+C" accumulates subtraction with
// no per-use negation. Diagonal and above hold U unmodified.
__global__ __launch_bounds__(256) void DistOA_solve_kernel(
    const float* __restrict__ C, const float* __restrict__ rm,
    const float* __restrict__ eps_param, const int* __restrict__ bad,
    float* __restrict__ ws_bd, float* __restrict__ ws_rowsum)
{
    extern __shared__ float As[];   // 256 x SS floats
    const int kb  = blockIdx.x;           // 0..511
    const int k   = kb >> 7;
    const int bb  = kb & 127;
    const int tid = threadIdx.x;

    const float rmk = sigmoidf_(rm[k]);
    const float eps = sigmoidf_(eps_param[k]);
    const int*   badrow = bad + bb * NB;
    const float* Cb     = C + (size_t)bb * NB * NPP;
    float* bg  = ws_bd     + (size_t)kb * NB * PB;
    float* rsg = ws_rowsum + (size_t)kb * NB;

    // ---------------- build phase: thread t owns column t ----------------
    const int badc = badrow[tid];
    for (int j = 0; j < NB; ++j) {
        __builtin_prefetch(&Cb[(j + 4) * NPP + tid], 0, 0);
        float cv = Cb[j * NPP + tid];
        float x  = rmk * cv;
        float cf = (1.0f + x) * __expf(-x);
        int   badj = badrow[j];
        float cm = ((badj != 0) == (badc != 0)) ? cf : 0.0f;
        if (tid == j) cm += eps;
        As[j * SS + tid] = cm;

        float bv = Cb[j * NPP + NB + tid];
        float xb = rmk * bv;
        float cfb = (1.0f + xb) * __expf(-xb);
        bg[j * PB + tid] = (badj != 0) ? 0.0f : cfb;
    }
    __syncthreads();
    // row sums of masked Cm (needed later for d1), before LU destroys As.
    // Staggered start column -> bank index (5*tid + c) mod 64, conflict-free.
    {
        float s = 0.0f;
        for (int c = 0; c < NB; ++c) s += As[tid * SS + ((c + tid) & (NB - 1))];
        rsg[tid] = s;
    }
    __syncthreads();

    const int wave = tid >> 5;
    const int lane = tid & 31;
    const int n    = lane & 15;
    const int hi   = lane >> 4;

    // ---------------- blocked LU (no pivoting), NB=16 panels ----------------
    for (int kk = 0; kk < NB; kk += 16) {
        // panel factor: columns kk..kk+15; store NEGATED multipliers below diag
        for (int j = kk; j < kk + 16; ++j) {
            __syncthreads();
            float rinv = 1.0f / As[j * SS + j];
            if (tid > j) {
                float mneg = -As[tid * SS + j] * rinv;       // -L(tid,j)
                As[tid * SS + j] = mneg;
                for (int c = j + 1; c < kk + 16; ++c)
                    As[tid * SS + c] = fmaf(mneg, As[j * SS + c], As[tid * SS + c]);
            }
        }
        __syncthreads();
        // U panel: L11^{-1} * A12 (unit-lower fwd subst; L stored negated -> FMA)
        {
            int c = kk + 16 + tid;
            if (c < NB) {
                for (int j = kk + 1; j < kk + 16; ++j) {
                    float s = As[j * SS + c];
                    for (int m = kk; m < j; ++m)
                        s = fmaf(As[j * SS + m], As[m * SS + c], s);  // +(-L)*U
                    As[j * SS + c] = s;
                }
            }
        }
        __syncthreads();
        // trailing update A22 += (-L21)*U12 via fp32 WMMA
        {
            int o  = kk + 16;
            int nt = (NB - o) >> 4;
            if (nt > 0) {
                int total = nt * nt;
                for (int t = wave; t < total; t += 8) {
                    int ti = t / nt, tj = t % nt;
                    int ro = o + ti * 16, co = o + tj * 16;
                    v8f cf;
                    #pragma unroll
                    for (int v = 0; v < 8; ++v)
                        cf[v] = As[(ro + v + 8 * hi) * SS + co + n];
                    #pragma unroll
                    for (int ks = 0; ks < 4; ++ks) {
                        v2f a, bf;
                        #pragma unroll
                        for (int v = 0; v < 2; ++v) {
                            int kr = ks * 4 + v + 2 * hi;
                            a[v]  = As[(ro + n) * SS + kk + kr];    // -L (pre-negated)
                            bf[v] = As[(kk + kr) * SS + co + n];    //  U
                        }
                        cf = __builtin_amdgcn_wmma_f32_16x16x4_f32(
                                 false, a, false, bf, (short)0, cf, false, false);
                    }
                    #pragma unroll
                    for (int v = 0; v < 8; ++v)
                        As[(ro + v + 8 * hi) * SS + co + n] = cf[v];
                }
            }
        }
        __syncthreads();
    }

    // ---------------- triangular solves: wave owns 16-col RHS chunks ----------------
    for (int chunk = wave; chunk < 16; chunk += 8) {
        int pc = chunk * 16;
        v8f y[16];
        #pragma unroll
        for (int bi = 0; bi < 16; ++bi)
            #pragma unroll
            for (int v = 0; v < 8; ++v)
                y[bi][v] = bg[(bi * 16 + v + 8 * hi) * PB + pc + n];

        // forward: (unit-lower L) y = b ; L stored negated -> pure accumulate
        #pragma unroll
        for (int bi = 0; bi < 16; ++bi) {
            #pragma unroll
            for (int bj = 0; bj < bi; ++bj) {
                float oth[8];
                #pragma unroll
                for (int v = 0; v < 8; ++v) oth[v] = __shfl_xor(y[bj][v], 16, 32);
                #pragma unroll
                for (int ks = 0; ks < 4; ++ks) {
                    v2f a, bf;
                    #pragma unroll
                    for (int v = 0; v < 2; ++v) {
                        int kr = ks * 4 + v + 2 * hi;
                        a[v] = As[(bi * 16 + n) * SS + bj * 16 + kr];   // -L
                        bf[v] = ((kr >> 3) == hi) ? y[bj][kr & 7] : oth[kr & 7];
                    }
                    y[bi] = __builtin_amdgcn_wmma_f32_16x16x4_f32(
                                false, a, false, bf, (short)0, y[bi], false, false);
                }
            }
            // 16x16 unit-lower diagonal solve, scalar per lane (both halves identical)
            float oth[8], y16[16];
            #pragma unroll
            for (int v = 0; v < 8; ++v) oth[v] = __shfl_xor(y[bi][v], 16, 32);
            #pragma unroll
            for (int r = 0; r < 16; ++r)
                y16[r] = ((r >> 3) == hi) ? y[bi][r & 7] : oth[r & 7];
            #pragma unroll
            for (int r = 1; r < 16; ++r) {
                float s = y16[r];
                for (int m = 0; m < r; ++m)
                    s = fmaf(As[(bi * 16 + r) * SS + bi * 16 + m], y16[m], s); // +(-L)*y
                y16[r] = s;
            }
            #pragma unroll
            for (int v = 0; v < 8; ++v) y[bi][v] = y16[8 * hi + v];
        }

        // backward: U d = y (U not pre-negated; negate A fragment here)
        #pragma unroll
        for (int bi = 15; bi >= 0; --bi) {
            #pragma unroll
            for (int bj = bi + 1; bj < 16; ++bj) {
                float oth[8];
                #pragma unroll
                for (int v = 0; v < 8; ++v) oth[v] = __shfl_xor(y[bj][v], 16, 32);
                #pragma unroll
                for (int ks = 0; ks < 4; ++ks) {
                    v2f a, bf;
                    #pragma unroll
                    for (int v = 0; v < 2; ++v) {
                        int kr = ks * 4 + v + 2 * hi;
                        a[v] = -As[(bi * 16 + n) * SS + bj * 16 + kr];  // -U
                        bf[v] = ((kr >> 3) == hi) ? y[bj][kr & 7] : oth[kr & 7];
                    }
                    y[bi] = __builtin_amdgcn_wmma_f32_16x16x4_f32(
                                false, a, false, bf, (short)0, y[bi], false, false);
                }
            }
            float oth[8], y16[16];
            #pragma unroll
            for (int v = 0; v < 8; ++v) oth[v] = __shfl_xor(y[bi][v], 16, 32);
            #pragma unroll
            for (int r = 0; r < 16; ++r)
                y16[r] = ((r >> 3) == hi) ? y[bi][r & 7] : oth[r & 7];
            #pragma unroll
            for (int r = 15; r >= 0; --r) {
                float s = y16[r];
                for (int m = r + 1; m < 16; ++m)
                    s -= As[(bi * 16 + r) * SS + bi * 16 + m] * y16[m];
                y16[r] = s / As[(bi * 16 + r) * SS + bi * 16 + r];
            }
            #pragma unroll
            for (int v = 0; v < 8; ++v) y[bi][v] = y16[8 * hi + v];
        }

        // store d over b (disjoint columns per wave -> safe)
        #pragma unroll
        for (int bi = 0; bi < 16; ++bi)
            #pragma unroll
            for (int v = 0; v < 8; ++v)
                bg[(bi * 16 + v + 8 * hi) * PB + pc + n] = y[bi][v];
    }
}

// One workgroup per (k,b): column reductions, blend, einsum -> out (B,P,K).
__global__ __launch_bounds__(256) void DistOA_out_kernel(
    const float* __restrict__ C, const float* __restrict__ val,
    const float* __restrict__ rm, const int* __restrict__ bad,
    const float* __restrict__ ws_bd, const float* __restrict__ ws_rowsum,
    float* __restrict__ out)
{
    __shared__ float rs[NB];
    const int kb = blockIdx.x;
    const int k  = kb >> 7;
    const int bb = kb & 127;
    const int t  = threadIdx.x;      // p index
    const float rmk = sigmoidf_(rm[k]);
    rs[t] = ws_rowsum[(size_t)kb * NB + t];
    __syncthreads();

    const float* Cb = C + (size_t)bb * NB * NPP;
    const float* dg = ws_bd + (size_t)kb * NB * PB;
    const int* badrow = bad + bb * NB;

    float sabs = 0.0f, s1 = 0.0f;
    for (int j = 0; j < NB; ++j) {
        float dv = dg[j * PB + t];
        sabs += fabsf(dv);
        float cv = Cb[j * NPP + NB + t];
        float x  = rmk * cv;
        float cf = (1.0f + x) * __expf(-x);
        float bv = (badrow[j] != 0) ? 0.0f : cf;
        s1 += bv / rs[j];
    }
    float wg = fminf(fmaxf((sabs - 1.0f) * 2.0f, 0.0f), 1.0f);
    float w2 = wg / fmaxf(s1, 1.0f);
    float omw = 1.0f - wg;

    float acc = 0.0f;
    for (int j = 0; j < NB; ++j) {
        float dv = dg[j * PB + t];
        float cv = Cb[j * NPP + NB + t];
        float x  = rmk * cv;
        float cf = (1.0f + x) * __expf(-x);
        float d1 = ((badrow[j] != 0) ? 0.0f : cf) / rs[j];
        float r  = dv * omw + d1 * w2;
        float vv = val[((size_t)bb * NB + j) * 4 + k];
        vv = (vv == vv) ? vv : 0.0f;                             // nan_to_num
        vv = fminf(fmaxf(vv, -3.4028235e38f), 3.4028235e38f);    // inf clamp
        acc += r * vv;
    }
    out[((size_t)bb * PB + t) * 4 + k] = acc;
}

extern "C" void kernel_launch(void* const* d_in, const int* in_sizes, int n_in,
                              void* d_out, int out_size, void* d_ws, size_t ws_size,
                              hipStream_t stream) {
    const float* C    = (const float*)d_in[0];   // (B,N,N+P)
    const float* val  = (const float*)d_in[1];   // (B,N,K)
    const float* rm   = (const float*)d_in[2];   // (K,1)
    const float* epsp = (const float*)d_in[3];   // (K,)
    const int*   bad  = (const int*)d_in[4];     // (B,N) bool -> int
    float* out = (float*)d_out;                  // (B,P,K)

    float* ws    = (float*)d_ws;
    float* ws_bd = ws;                                   // 512 * 256 * 256 floats (b, then d)
    float* ws_rs = ws + (size_t)512 * NB * PB;           // 512 * 256 floats (rowsums)

    const size_t ldsBytes = (size_t)NB * SS * sizeof(float);  // ~260 KB of the 320 KB LDS
    DistOA_solve_kernel<<<dim3(512), dim3(256), ldsBytes, stream>>>(
        C, rm, epsp, bad, ws_bd, ws_rs);
    DistOA_out_kernel<<<dim3(512), dim3(256), 0, stream>>>(
        C, val, rm, bad, ws_bd, ws_rs, out);
}